// DeconvolutionEnergyScoreLoss_21157008900184
// MI455X (gfx1250) — compile-verified
//
#include <hip/hip_runtime.h>
#include <hip/hip_bf16.h>
#include <math.h>

// Problem constants (match reference)
#define Bn   64
#define Gx   32
#define Dd   2000
#define Mm   16
#define Ee   16
#define DINc 256
#define KK   272      // DIN + E
#define NT_COLT 125   // Dd / 16

typedef __attribute__((ext_vector_type(2))) float v2f;
typedef __attribute__((ext_vector_type(8))) float v8f;

// ---------------------------------------------------------------------------
// V_WMMA_F32_16X16X4_F32 helpers.
// A: row-major [16 x K] (LDS), lda leading dim.
// B: row-major [K x ldb] (global), columns [n0, n0+16).
// A frag: lanes 0-15 = {K+0,K+1}, lanes 16-31 = {K+2,K+3}, row = lane&15
// B frag: same K split, col = n0 + (lane&15)
// C/D:    acc[i] -> row (i + 8*(lane>=16)), col (n0 + lane&15)
// ---------------------------------------------------------------------------
__device__ __forceinline__ v8f wmma_f32(v2f a, v2f b, v8f c) {
#if defined(__HIP_DEVICE_COMPILE__)
  return __builtin_amdgcn_wmma_f32_16x16x4_f32(false, a, false, b,
                                               (short)0, c, false, false);
#else
  for (int i = 0; i < 8; ++i) c[i] += a.x * b.x + a.y * b.y;  // host stub
  return c;
#endif
}

__device__ __forceinline__ v2f load_bfrag(const float* __restrict__ bcol, int ldb, int k) {
  v2f b;
  b.x = bcol[(size_t)k * ldb];
  b.y = bcol[(size_t)(k + 1) * ldb];
  return b;
}

// Single 16x16 output tile (software-pipelined B loads).
__device__ __forceinline__ v8f wmma_tile_f32(const float* __restrict__ As, int lda,
                                             const float* __restrict__ Bg, int ldb,
                                             int n0, int K) {
  const int lane = threadIdx.x & 31;
  const int half = lane >> 4;
  const int l16  = lane & 15;
  const float* arow = As + l16 * lda + (half << 1);
  const float* bcol = Bg + (size_t)(half << 1) * ldb + n0 + l16;
  v8f acc = {0.f, 0.f, 0.f, 0.f, 0.f, 0.f, 0.f, 0.f};
  v2f bc = load_bfrag(bcol, ldb, 0);
  for (int k = 0; k < K; k += 4) {
    const int kn = (k + 4 < K) ? (k + 4) : k;   // clamp: branch-free pipeline
    v2f bn = load_bfrag(bcol, ldb, kn);
    v2f a;
    a.x = arow[k];
    a.y = arow[k + 1];
    acc = wmma_f32(a, bc, acc);
    bc = bn;
  }
  return acc;
}

// Two row-tiles sharing one B stream (B fragment consumed by 2 WMMAs).
__device__ __forceinline__ void wmma_tile2_f32(const float* __restrict__ As0,
                                               const float* __restrict__ As1, int lda,
                                               const float* __restrict__ Bg, int ldb,
                                               int n0, int K, v8f& acc0, v8f& acc1) {
  const int lane = threadIdx.x & 31;
  const int half = lane >> 4;
  const int l16  = lane & 15;
  const float* a0 = As0 + l16 * lda + (half << 1);
  const float* a1 = As1 + l16 * lda + (half << 1);
  const float* bcol = Bg + (size_t)(half << 1) * ldb + n0 + l16;
  v2f bc = load_bfrag(bcol, ldb, 0);
  for (int k = 0; k < K; k += 4) {
    const int kn = (k + 4 < K) ? (k + 4) : k;
    v2f bn = load_bfrag(bcol, ldb, kn);
    v2f x0, x1;
    x0.x = a0[k]; x0.y = a0[k + 1];
    x1.x = a1[k]; x1.y = a1[k + 1];
    acc0 = wmma_f32(x0, bc, acc0);
    acc1 = wmma_f32(x1, bc, acc1);
    bc = bn;
  }
}

__device__ __forceinline__ float wave_sum32(float v) {
  for (int off = 16; off > 0; off >>= 1) v += __shfl_xor(v, off, 32);
  return v;
}

__device__ __forceinline__ float softplusf(float z) {
  // jax.nn.softplus: max(z,0) + log1p(exp(-|z|))
  return fmaxf(z, 0.f) + log1pf(expf(-fabsf(z)));
}

// ---------------------------------------------------------------------------
// K1: A1[b*16+m][0:256] = sum_g x[b,g,:];  A1[b*16+m][256+e] = sum_g noise[b,g,m,e]
// ---------------------------------------------------------------------------
__global__ __launch_bounds__(256)
void build_a1_kernel(const float* __restrict__ x, const float* __restrict__ noise_loss,
                     float* __restrict__ A1) {
  const int b = blockIdx.x, t = threadIdx.x;
  float s = 0.f;
  for (int g = 0; g < Gx; ++g) s += x[(size_t)(b * Gx + g) * DINc + t];
  for (int m = 0; m < Mm; ++m) A1[(size_t)(b * Mm + m) * KK + t] = s;
  const int m = t >> 4, e = t & 15;
  float sn = 0.f;
  for (int g = 0; g < Gx; ++g)
    sn += noise_loss[(size_t)((b * Gx + g) * Mm + m) * Ee + e];
  A1[(size_t)(b * Mm + m) * KK + DINc + e] = sn;
}

// ---------------------------------------------------------------------------
// K2: per-b loss partial. block 256 (8 waves), dyn LDS ~148.5 KB
// ---------------------------------------------------------------------------
__global__ __launch_bounds__(256)
void loss_kernel(const float* __restrict__ A1, const float* __restrict__ W,
                 const float* __restrict__ bias, const float* __restrict__ target,
                 float* __restrict__ lpart) {
  extern __shared__ float sm[];
  float* Pb  = sm;                // 16*2000
  float* As  = Pb + Mm * Dd;      // 16*272
  float* inn = As + Mm * KK;      // 256
  float* red = inn + 256;         // 256
  float* tc  = red + 256;         // 16
  float* pw  = tc + 16;           // 256

  const int b = blockIdx.x, tid = threadIdx.x;
  for (int i = tid; i < Mm * KK; i += 256) As[i] = A1[(size_t)b * Mm * KK + i];
  __syncthreads();

  const int wave = tid >> 5, lane = tid & 31, half = lane >> 4, l16 = lane & 15;
  for (int ct = wave; ct < NT_COLT; ct += 8) {
    const int n0 = ct * 16;
    v8f acc = wmma_tile_f32(As, KK, W, Dd, n0, KK);
    for (int i = 0; i < 8; ++i) {
      const int row = half * 8 + i, col = n0 + l16;
      Pb[row * Dd + col] = acc[i] + 32.f * bias[col];   // + G*bias
    }
  }
  __syncthreads();

  // inn[m][n] = sum_d Pb[m]·Pb[n]; term_conf partials
  {
    const int m = tid >> 4, n = tid & 15;
    float s = 0.f;
    for (int d = 0; d < Dd; ++d) s += Pb[m * Dd + d] * Pb[n * Dd + d];
    inn[tid] = s;
    float s2 = 0.f;
    for (int d = n; d < Dd; d += 16) {
      const float v = Pb[m * Dd + d] - target[(size_t)b * Dd + d];
      s2 += v * v;
    }
    red[tid] = s2;
  }
  __syncthreads();

  if (tid < Mm) {
    float s = 0.f;
    for (int i = 0; i < 16; ++i) s += red[tid * 16 + i];
    tc[tid] = sqrtf(s);
  }
  {
    const int m = tid >> 4, n = tid & 15;
    float dv = 0.f;
    if (m != n) {
      const float sqd = inn[m * 16 + m] + inn[n * 16 + n] - 2.f * inn[tid];
      dv = sqrtf(fmaxf(sqd, 1e-6f));
    }
    pw[tid] = dv;
  }
  __syncthreads();

  if (tid == 0) {
    float tcs = 0.f;
    for (int i = 0; i < 16; ++i) tcs += tc[i];
    float pws = 0.f;
    for (int i = 0; i < 256; ++i) pws += pw[i];
    const float term_conf = tcs * (1.f / 16.f);
    const float mean_pd = pws * (1.f / (16.f * 15.f));
    lpart[b] = (term_conf - 0.5f * mean_pd) * (1.f / (float)Bn);
  }
}

__global__ __launch_bounds__(64)
void finalize_loss_kernel(const float* __restrict__ lpart, float* __restrict__ out) {
  __shared__ float sh[64];
  sh[threadIdx.x] = lpart[threadIdx.x];
  __syncthreads();
  if (threadIdx.x == 0) {
    float s = 0.f;
    for (int i = 0; i < 64; ++i) s += sh[i];
    out[0] = s;
  }
}

// ---------------------------------------------------------------------------
// K4: per-b fused  GEMM+softplus -> IPF(60) -> integer rounding.
// block 1024 (32 waves), dyn LDS = (32*2000 + 32*272)*4 = 290,816 B
// ---------------------------------------------------------------------------
__global__ __launch_bounds__(1024)
void sample_ipf_round_kernel(const float* __restrict__ x, const int* __restrict__ tsum,
                             const float* __restrict__ W, const float* __restrict__ bias,
                             const float* __restrict__ noise, int* __restrict__ outI) {
  extern __shared__ float sm[];
  float* y  = sm;                 // 32*2000
  float* As = sm + Gx * Dd;       // 32*272  (reused after GEMM)
  float* Cb = As;                 // 2000
  float* Rb = Cb + Dd;            // 32
  float* cs = Rb + Gx;            // 2000

  const int b = blockIdx.x, tid = threadIdx.x;
  const int wave = tid >> 5, lane = tid & 31, half = lane >> 4, l16 = lane & 15;

  // stage A = concat(x_rows, noise_rows) for this b
  for (int i = tid; i < Gx * DINc; i += 1024) {
    const int g = i / DINc, k = i - g * DINc;
    As[g * KK + k] = x[(size_t)(b * Gx + g) * DINc + k];
  }
  for (int i = tid; i < Gx * Ee; i += 1024) {
    const int g = i >> 4, e = i & 15;
    As[g * KK + DINc + e] = noise[(size_t)(b * Gx + g) * Ee + e];
  }
  __syncthreads();

  // GEMM (32 x 2000, K=272): wave takes one column tile, computes BOTH row
  // tiles with a shared B stream (each B fragment feeds two WMMAs).
  for (int ct = wave; ct < NT_COLT; ct += 32) {
    const int n0 = ct * 16;
    v8f acc0 = {0.f, 0.f, 0.f, 0.f, 0.f, 0.f, 0.f, 0.f};
    v8f acc1 = {0.f, 0.f, 0.f, 0.f, 0.f, 0.f, 0.f, 0.f};
    wmma_tile2_f32(As, As + 16 * KK, KK, W, Dd, n0, KK, acc0, acc1);
    for (int i = 0; i < 8; ++i) {
      const int row = half * 8 + i, col = n0 + l16;
      const float bb = bias[col];
      y[row * Dd + col]        = softplusf(acc0[i] + bb);
      y[(row + 16) * Dd + col] = softplusf(acc1[i] + bb);
    }
  }
  __syncthreads();

  // IPF setup: C, anchor row totals R, zero-smoothing
  for (int d = tid; d < Dd; d += 1024) Cb[d] = fmaxf((float)tsum[(size_t)b * Dd + d], 0.f);
  {
    float s = 0.f;
    for (int d = lane; d < Dd; d += 32) s += y[wave * Dd + d];
    s = wave_sum32(s);
    if (lane == 0) Rb[wave] = s;
  }
  __syncthreads();
  for (int d = tid; d < Dd; d += 1024) {
    float s = 0.f;
    for (int g = 0; g < Gx; ++g) s += y[g * Dd + d];
    if (s <= 0.f && Cb[d] > 0.f)
      for (int g = 0; g < Gx; ++g) y[g * Dd + d] += 1e-8f;
  }
  __syncthreads();
  {
    float s = 0.f;
    for (int d = lane; d < Dd; d += 32) s += y[wave * Dd + d];
    s = wave_sum32(s);
    if (s <= 0.f && Rb[wave] > 0.f)
      for (int d = lane; d < Dd; d += 32) y[wave * Dd + d] += 1e-8f;
  }
  __syncthreads();

  // 60 IPF iterations, y resident in LDS
  for (int it = 0; it < 60; ++it) {
    for (int d = tid; d < Dd; d += 1024) {
      float s = 0.f;
      for (int g = 0; g < Gx; ++g) s += y[g * Dd + d];
      float f = Cb[d] / fmaxf(s, 1e-12f);
      cs[d] = fminf(fmaxf(f, 0.75f), 1.25f);
    }
    __syncthreads();
    {
      // apply col scale + row sum + row scale (wave owns its row)
      float s = 0.f;
      for (int d = lane; d < Dd; d += 32) {
        const float v = y[wave * Dd + d] * cs[d];
        y[wave * Dd + d] = v;
        s += v;
      }
      s = wave_sum32(s);
      float f = Rb[wave] / fmaxf(s, 1e-12f);
      f = fminf(fmaxf(f, 0.75f), 1.25f);
      for (int d = lane; d < Dd; d += 32) y[wave * Dd + d] *= f;
    }
    __syncthreads();
  }

  // final column normalization (unclipped)
  for (int d = tid; d < Dd; d += 1024) {
    float s = 0.f;
    for (int g = 0; g < Gx; ++g) s += y[g * Dd + d];
    cs[d] = Cb[d] / fmaxf(s, 1e-12f);
  }
  __syncthreads();
  for (int g = 0; g < Gx; ++g)
    for (int d = tid; d < Dd; d += 1024)
      y[g * Dd + d] = fmaxf(y[g * Dd + d] * cs[d], 0.f);
  __syncthreads();

  // exact integerization per column (stable double-argsort rank semantics)
  for (int d = tid; d < Dd; d += 1024) {
    const int t = tsum[(size_t)b * Dd + d];
    int s0 = 0;
    for (int g = 0; g < Gx; ++g) s0 += (int)floorf(y[g * Dd + d]);
    const int need = t - s0;
    const int pos = need > 0 ? need : 0;
    const int q = pos >> 5;
    const int r = pos & 31;

    unsigned addmask = 0u;
    if (r > 0) {
      for (int g = 0; g < Gx; ++g) {
        const float vg = y[g * Dd + d];
        const float fg = vg - floorf(vg);
        int rank = 0;
        for (int g2 = 0; g2 < Gx; ++g2) {
          const float v2 = y[g2 * Dd + d];
          const float f2 = v2 - floorf(v2);
          rank += (f2 > fg) || (f2 == fg && g2 < g);
        }
        if (rank < r) addmask |= (1u << g);
      }
    }

    const int s1 = s0 + (q << 5) + r;
    const int need2 = t - s1;
    int neg = need2 < 0 ? -need2 : 0;
    neg = neg < s1 ? neg : s1;
    const int q2 = neg >> 5;

    int removed = 0;
    for (int g = 0; g < Gx; ++g) {
      const int yi = (int)floorf(y[g * Dd + d]) + q + (int)((addmask >> g) & 1u);
      removed += yi < q2 ? yi : q2;
    }
    const int r2 = neg - removed;

    unsigned remmask = 0u;
    if (r2 > 0) {
      for (int g = 0; g < Gx; ++g) {
        const float vg = y[g * Dd + d];
        const float fg = vg - floorf(vg);
        int yi2g = ((int)floorf(vg) + q + (int)((addmask >> g) & 1u)) - q2;
        if (yi2g < 0) yi2g = 0;
        const bool posg = yi2g > 0;
        int rank = 0;
        for (int g2 = 0; g2 < Gx; ++g2) {
          const float v2 = y[g2 * Dd + d];
          const float f2 = v2 - floorf(v2);
          int yi2b = ((int)floorf(v2) + q + (int)((addmask >> g2) & 1u)) - q2;
          const bool pos2 = yi2b > 0;
          bool lt, eq;
          if (pos2 && posg)       { lt = f2 < fg; eq = f2 == fg; }
          else if (pos2 && !posg) { lt = true;  eq = false; }   // finite < inf
          else if (!pos2 && posg) { lt = false; eq = false; }
          else                    { lt = false; eq = true;  }   // inf == inf
          rank += lt || (eq && g2 < g);
        }
        if (rank < r2) remmask |= (1u << g);
      }
    }

    for (int g = 0; g < Gx; ++g) {
      const float vg = y[g * Dd + d];
      int yi2g = ((int)floorf(vg) + q + (int)((addmask >> g) & 1u)) - q2;
      if (yi2g < 0) yi2g = 0;
      int o = yi2g - (int)((remmask >> g) & 1u);
      if (o < 0) o = 0;
      outI[(size_t)(b * Gx + g) * Dd + d] = o;
    }
  }
}

// ---------------------------------------------------------------------------
extern "C" void kernel_launch(void* const* d_in, const int* in_sizes, int n_in,
                              void* d_out, int out_size, void* d_ws, size_t ws_size,
                              hipStream_t stream) {
  const float* x            = (const float*)d_in[0];   // 2048 x 256
  const float* target       = (const float*)d_in[1];   // 64 x 2000
  const int*   target_sum   = (const int*)  d_in[2];   // 64 x 2000
  const float* W            = (const float*)d_in[3];   // 272 x 2000
  const float* bias         = (const float*)d_in[4];   // 2000
  const float* noise_loss   = (const float*)d_in[5];   // 32768 x 16
  const float* noise_sample = (const float*)d_in[6];   // 2048 x 16

  float* loss_out = (float*)d_out;          // output[0] = loss scalar
  int*   y_out    = (int*)d_out + 1;        // output[1:] = 2048 x 2000 int32

  float* A1    = (float*)d_ws;                          // 1024*272 floats
  float* lpart = A1 + (size_t)Bn * Mm * KK;             // 64 floats

  const size_t smem_loss   = (size_t)(Mm * Dd + Mm * KK + 256 + 256 + 16 + 256) * sizeof(float);
  const size_t smem_sample = (size_t)(Gx * Dd + Gx * KK) * sizeof(float);

  build_a1_kernel<<<Bn, 256, 0, stream>>>(x, noise_loss, A1);
  loss_kernel<<<Bn, 256, smem_loss, stream>>>(A1, W, bias, target, lpart);
  finalize_loss_kernel<<<1, 64, 0, stream>>>(lpart, loss_out);
  sample_ipf_round_kernel<<<Bn, 1024, smem_sample, stream>>>(x, target_sum, W, bias,
                                                             noise_sample, y_out);
}